// CinLayer_15487652069576
// MI455X (gfx1250) — compile-verified
//
#include <hip/hip_runtime.h>
#include <hip/hip_bf16.h>

typedef __attribute__((ext_vector_type(2))) float v2f;
typedef __attribute__((ext_vector_type(8))) float v8f;

// Problem constants (from reference setup_inputs).
#define BS   32
#define TS0  64
#define TSK  64
#define F    256
#define NF   128

// ---------------------------------------------------------------------------
// Stage 1: one block per fc in [0,256).
//   A[b, n, fc] = sum_y w[n, fc/4, y] * x_k[b, y, fc]
// GEMM per fc: M = NF = 128 (n), N = BS = 32 (b), K = 64 (y).
// Wave w handles m-tile = wave_id (8 tiles), loops over 2 n-tiles.
// ---------------------------------------------------------------------------
__global__ void cin_stage1_wmma(const float* __restrict__ xk,
                                const float* __restrict__ w,
                                float* __restrict__ A) {
    const int fc   = blockIdx.x;          // 0..255
    const int t    = fc >> 2;             // conv-kernel row this fc uses
    const int wave = threadIdx.x >> 5;    // 0..7  -> m-tile
    const int lane = threadIdx.x & 31;
    const int half = lane >> 4;           // 0: K pair {0,1}, 1: K pair {2,3}
    const int l    = lane & 15;

    const float* wrow = w + (size_t)t * TSK;            // w[., t, .], stride F? no: n stride = TS0*TSK
    (void)wrow;

    for (int nt = 0; nt < 2; ++nt) {
        v8f acc = {};
        #pragma unroll
        for (int kc = 0; kc < 16; ++kc) {
            const int y0 = kc * 4 + half * 2;           // K offset for this half-wave
            // A-frag: W[m = mt*16+l, k = y0, y0+1]
            const int n = wave * 16 + l;
            v2f a;
            a.x = w[(size_t)n * (TS0 * TSK) + (size_t)t * TSK + y0];
            a.y = w[(size_t)n * (TS0 * TSK) + (size_t)t * TSK + y0 + 1];
            // B-frag: Xk[k = y0, y0+1, ncol = nt*16+l]  (ncol indexes batch b)
            const int b = nt * 16 + l;
            v2f bb;
            bb.x = xk[(size_t)b * (TSK * F) + (size_t)y0 * F + fc];
            bb.y = xk[(size_t)b * (TSK * F) + (size_t)(y0 + 1) * F + fc];
            acc = __builtin_amdgcn_wmma_f32_16x16x4_f32(
                /*neg_a=*/false, a, /*neg_b=*/false, bb,
                /*c_mod=*/(short)0, acc, /*reuse_a=*/false, /*reuse_b=*/false);
        }
        // D tile: VGPR v, lanes 0-15 -> M=v, lanes 16-31 -> M=v+8, N=lane&15.
        #pragma unroll
        for (int v = 0; v < 8; ++v) {
            const int n = wave * 16 + v + half * 8;
            const int b = nt * 16 + l;
            A[((size_t)b * NF + n) * F + fc] = acc[v];
        }
    }
}

// ---------------------------------------------------------------------------
// Stage 2: one block per (b, q), q in [0,4).
//   out[b, n, 64q + r] = sum_t A[b, n, 4t+q] * x_0[b, r, 4t+q] + bias[n]
// GEMM: M = NF = 128 (n), N = 64 (r), K = 64 (t).
// Wave w handles m-tile = wave_id (8 tiles), loops over 4 r-tiles.
// ---------------------------------------------------------------------------
__global__ void cin_stage2_wmma(const float* __restrict__ x0,
                                const float* __restrict__ A,
                                const float* __restrict__ bias,
                                float* __restrict__ out) {
    const int bq   = blockIdx.x;          // 0..127
    const int b    = bq >> 2;
    const int q    = bq & 3;
    const int wave = threadIdx.x >> 5;    // m-tile
    const int lane = threadIdx.x & 31;
    const int half = lane >> 4;
    const int l    = lane & 15;

    for (int rt = 0; rt < 4; ++rt) {
        v8f acc = {};
        #pragma unroll
        for (int kc = 0; kc < 16; ++kc) {
            const int t0 = kc * 4 + half * 2;           // K (= t) offset for this half-wave
            // A-frag: Aq[m = n, k = t0, t0+1] = A[b, n, 4t + q]
            const int n = wave * 16 + l;
            v2f a;
            a.x = A[((size_t)b * NF + n) * F + (4 * t0 + q)];
            a.y = A[((size_t)b * NF + n) * F + (4 * (t0 + 1) + q)];
            // B-frag: X0q[k = t0, t0+1, ncol = r] = x_0[b, r, 4t + q]
            const int r = rt * 16 + l;
            v2f bb;
            bb.x = x0[(size_t)b * (TS0 * F) + (size_t)r * F + (4 * t0 + q)];
            bb.y = x0[(size_t)b * (TS0 * F) + (size_t)r * F + (4 * (t0 + 1) + q)];
            acc = __builtin_amdgcn_wmma_f32_16x16x4_f32(
                false, a, false, bb, (short)0, acc, false, false);
        }
        #pragma unroll
        for (int v = 0; v < 8; ++v) {
            const int n = wave * 16 + v + half * 8;
            const int r = rt * 16 + l;
            out[((size_t)b * NF + n) * F + q * 64 + r] = acc[v] + bias[n];
        }
    }
}

// ---------------------------------------------------------------------------
// Launch: two dependent GEMM stages on the same stream; intermediate A
// (32*128*256 floats = 4 MB) lives in d_ws and is fully overwritten by
// stage 1 before stage 2 reads it (deterministic, no cross-call state).
// ---------------------------------------------------------------------------
extern "C" void kernel_launch(void* const* d_in, const int* in_sizes, int n_in,
                              void* d_out, int out_size, void* d_ws, size_t ws_size,
                              hipStream_t stream) {
    const float* x0   = (const float*)d_in[0];   // (32, 64, 256)
    const float* xk   = (const float*)d_in[1];   // (32, 64, 256)
    const float* w    = (const float*)d_in[2];   // (128, 64, 64)
    const float* bias = (const float*)d_in[3];   // (128,)
    float* A   = (float*)d_ws;                   // (32, 128, 256) intermediate
    float* out = (float*)d_out;                  // (32, 128, 256)

    cin_stage1_wmma<<<F, 256, 0, stream>>>(xk, w, A);
    cin_stage2_wmma<<<BS * 4, 256, 0, stream>>>(x0, A, bias, out);
}